// NeuralODEfunc_4569845203188
// MI455X (gfx1250) — compile-verified
//
#include <hip/hip_runtime.h>
#include <math.h>

#define RES   96
#define RES2  (RES*RES)          // 9216
#define NPTS  (RES*RES*RES)      // 884736
#define PTS_PER_BLOCK 128        // 8 waves * 16 points
#define OUT_PER_BLOCK (PTS_PER_BLOCK*27)  // 3456 floats
#define OUT4_PER_BLOCK (OUT_PER_BLOCK/4)  // 864 float4

typedef float v2f __attribute__((ext_vector_type(2)));
typedef float v8f __attribute__((ext_vector_type(8)));

// ---- Brill-Lindquist conformal factor^4:  F = (1 + sum m/(2r))^4 ----
// meshgrid indexing='xy': X = xg[i1], Y = xg[i0], Z = xg[i2]
__device__ __forceinline__ float bl_F(int i0, int i1, int i2,
                                      const float* __restrict__ pos,
                                      float m0, float m1) {
    const float DXc  = 1.0f / 47.0f;                 // MAX_X/(RES/2-1)
    const float x0   = -47.0f * DXc;                 // linspace start
    const float step = (94.0f * DXc) / 95.0f;        // linspace step (96 pts)
    float X = x0 + step * (float)i1;
    float Y = x0 + step * (float)i0;
    float Z = x0 + step * (float)i2;
    float psi = 1.0f;
    {
        float dx = X - pos[0], dy = Y - pos[1], dz = Z - pos[2];
        float r = sqrtf(dx*dx + dy*dy + dz*dz);
        psi += m0 / (2.0f * r);
    }
    {
        float dx = X - pos[3], dy = Y - pos[4], dz = Z - pos[5];
        float r = sqrtf(dx*dx + dy*dy + dz*dz);
        psi += m1 / (2.0f * r);
    }
    float p2 = psi * psi;
    return p2 * p2;
}

// ---- Pass 1: F = psi^4 over the grid (3.5 MB, L2-resident) ----
__global__ void __launch_bounds__(256)
k_psi4(const float* __restrict__ pos, const float* __restrict__ mpre,
       float* __restrict__ Fb) {
    int p = blockIdx.x * blockDim.x + threadIdx.x;
    if (p >= NPTS) return;
    float m0 = log1pf(expf(mpre[0]));   // softplus
    float m1 = log1pf(expf(mpre[1]));
    int i2 = p % RES;
    int i1 = (p / RES) % RES;
    int i0 = p / RES2;
    Fb[p] = bl_F(i0, i1, i2, pos, m0, m1);
}

__device__ __forceinline__ float getF(const float* __restrict__ Fb,
                                      int i0, int i1, int i2,
                                      const float* __restrict__ pos,
                                      float m0, float m1, int useF) {
    if (useF) return Fb[(i0 * RES + i1) * RES + i2];
    return bl_F(i0, i1, i2, pos, m0, m1);
}

// Coefficient of u_K in Gamma component c = i*9+j*3+k:
//   d_ij*[k==K] + d_ik*[j==K] - d_jk*[i==K]
__device__ __forceinline__ float gcoef(int c, int K) {
    if (c >= 27 || K >= 3) return 0.0f;
    int i = c / 9, j = (c / 3) % 3, k = c % 3;
    float v = 0.0f;
    if (i == j && k == K) v += 1.0f;
    if (i == k && j == K) v += 1.0f;
    if (j == k && i == K) v -= 1.0f;
    return v;
}

// ---- Pass 2: finite-difference u, WMMA 3->27 expansion, coalesced store ----
__global__ void __launch_bounds__(256)
k_gamma(const float* __restrict__ Fb, const float* __restrict__ pos,
        const float* __restrict__ mpre, float* __restrict__ out, int useF) {
    __shared__ __align__(16) float lds[OUT_PER_BLOCK];

    const float inv_dx = 47.0f;   // 1/DX (gradient spacing is DX)

    int tid  = threadIdx.x;
    int wave = tid >> 5;
    int lane = tid & 31;
    int nlo  = lane & 15;
    int hi   = lane >> 4;

    float m0 = 0.0f, m1 = 0.0f;
    if (!useF) { m0 = log1pf(expf(mpre[0])); m1 = log1pf(expf(mpre[1])); }

    // Lanes 16..31 recompute the same point as lanes 0..15: they supply the
    // K=2 (u_z) half of the WMMA A-operand without any cross-lane shuffle.
    int p  = blockIdx.x * PTS_PER_BLOCK + wave * 16 + nlo;
    int i2 = p % RES;
    int i1 = (p / RES) % RES;
    int i0 = p / RES2;

    float Fc    = getF(Fb, i0, i1, i2, pos, m0, m1, useF);
    float inv2F = 0.5f / Fc;

    float u[3];
    {   // array axis 0
        int tm = (i0 > 0) ? i0 - 1 : 0;
        int tp = (i0 < RES - 1) ? i0 + 1 : RES - 1;
        float s = (tp - tm == 2) ? 0.5f * inv_dx : inv_dx;
        u[0] = (getF(Fb, tp, i1, i2, pos, m0, m1, useF) -
                getF(Fb, tm, i1, i2, pos, m0, m1, useF)) * s * inv2F;
    }
    {   // array axis 1
        int tm = (i1 > 0) ? i1 - 1 : 0;
        int tp = (i1 < RES - 1) ? i1 + 1 : RES - 1;
        float s = (tp - tm == 2) ? 0.5f * inv_dx : inv_dx;
        u[1] = (getF(Fb, i0, tp, i2, pos, m0, m1, useF) -
                getF(Fb, i0, tm, i2, pos, m0, m1, useF)) * s * inv2F;
    }
    {   // array axis 2
        int tm = (i2 > 0) ? i2 - 1 : 0;
        int tp = (i2 < RES - 1) ? i2 + 1 : RES - 1;
        float s = (tp - tm == 2) ? 0.5f * inv_dx : inv_dx;
        u[2] = (getF(Fb, i0, i1, tp, pos, m0, m1, useF) -
                getF(Fb, i0, i1, tm, pos, m0, m1, useF)) * s * inv2F;
    }

    // A (16x4 f32): VGPR0 = {K0 | K2} across lane halves, VGPR1 = {K1 | K3}
    v2f A;
    A[0] = hi ? u[2] : u[0];
    A[1] = hi ? 0.0f : u[1];

    // B (4x16 f32), dual layout: VGPR0 = {K0 | K2}, VGPR1 = {K1 | K3}
    int Klo = hi ? 2 : 0;
    int Khi = hi ? 3 : 1;
    v2f B0, B1;
    B0[0] = gcoef(nlo, Klo);        B0[1] = gcoef(nlo, Khi);       // comps 0..15
    B1[0] = gcoef(16 + nlo, Klo);   B1[1] = gcoef(16 + nlo, Khi);  // comps 16..26

    v8f C = {0.f,0.f,0.f,0.f,0.f,0.f,0.f,0.f};
    // D[M=point][N=component] = sum_K u_K * coef(N,K)
    v8f D0 = __builtin_amdgcn_wmma_f32_16x16x4_f32(
        false, A, false, B0, (short)0, C, false, false);
    v8f D1 = __builtin_amdgcn_wmma_f32_16x16x4_f32(
        false, A, false, B1, (short)0, C, false, false);

    // D layout: VGPR v holds row M=v (lanes 0-15) / M=v+8 (lanes 16-31), N=nlo
    int rowBase = wave * 16;
#pragma unroll
    for (int v = 0; v < 8; ++v) {
        int M = v + 8 * hi;
        lds[(rowBase + M) * 27 + nlo] = D0[v];
    }
    if (nlo < 11) {
#pragma unroll
        for (int v = 0; v < 8; ++v) {
            int M = v + 8 * hi;
            lds[(rowBase + M) * 27 + 16 + nlo] = D1[v];
        }
    }
    __syncthreads();

    // Fully coalesced streaming store: 864 float4 (b128) per block
    const float4* lds4 = (const float4*)lds;
    float4* out4 = (float4*)(out + (size_t)blockIdx.x * OUT_PER_BLOCK);
    for (int s = tid; s < OUT4_PER_BLOCK; s += 256)
        out4[s] = lds4[s];
}

extern "C" void kernel_launch(void* const* d_in, const int* in_sizes, int n_in,
                              void* d_out, int out_size, void* d_ws, size_t ws_size,
                              hipStream_t stream) {
    const float* pos  = (const float*)d_in[0];   // [2,3]
    const float* mpre = (const float*)d_in[1];   // [2]
    float* out = (float*)d_out;                  // [96,96,96,3,3,3]
    float* Fb  = (float*)d_ws;

    int useF = (ws_size >= (size_t)NPTS * sizeof(float)) ? 1 : 0;
    if (useF) {
        k_psi4<<<NPTS / 256, 256, 0, stream>>>(pos, mpre, Fb);
    }
    k_gamma<<<NPTS / PTS_PER_BLOCK, 256, 0, stream>>>(Fb, pos, mpre, out, useF);
}